// Block_66795331387589
// MI455X (gfx1250) — compile-verified
//
#include <hip/hip_runtime.h>

typedef __attribute__((ext_vector_type(2))) float v2f;
typedef __attribute__((ext_vector_type(8))) float v8f;

#define SDIM 32
#define SP 32768              // 32^3
#define CIN_SV 84
#define CIN_T 144
#define CIN 228
#define COUT 84
#define COUT_PAD 96
#define TAPS 343
#define TAPS_PK 392           // 7*7*8 : kw padded to 8 (kw==7 -> zero weight)
#define NBATCH 2
#define TW 40                 // staged halo row: 38 valid + 2 zero pad
#define TILE_N (7 * 7 * TW)   // 1960 floats
#define NSTAGE 11             // ceil(TILE_N / 192)

// ---------------------------------------------------------------------------
// Kernel 0: init the 4 per-block norm^2 maxima (deterministic across replays)
// ---------------------------------------------------------------------------
__global__ void init_factors(unsigned* __restrict__ f) {
    if (threadIdx.x < 4) f[threadIdx.x] = 0u;
}

// ---------------------------------------------------------------------------
// Kernel 1: tensor-product channels t = v_i v_j + per-block max norm^2
// ---------------------------------------------------------------------------
__global__ void tensor_norm_kernel(const float* __restrict__ sv5,
                                   float* __restrict__ t,
                                   unsigned* __restrict__ factors) {
    int idx = blockIdx.x * blockDim.x + threadIdx.x;   // 65536 total
    int b   = idx >> 15;
    int pos = idx & (SP - 1);
    const float* x = sv5 + (size_t)b * CIN_SV * SP + pos;

    float mx0 = 0.f, mx1 = 0.f, mx2 = 0.f, mx3 = 0.f;

    for (int m = 0; m < 16; ++m) {                 // block 0: (16,1)
        float v = x[(size_t)m * SP];
        mx0 = fmaxf(mx0, v * v);
    }
    float* tp = t + (size_t)b * CIN_T * SP + pos;  // block 1: (16,3) + t
    for (int m = 0; m < 16; ++m) {
        float v0 = x[(size_t)(16 + 3 * m + 0) * SP];
        float v1 = x[(size_t)(16 + 3 * m + 1) * SP];
        float v2 = x[(size_t)(16 + 3 * m + 2) * SP];
        float s = v0 * v0 + v1 * v1 + v2 * v2;
        mx1 = fmaxf(mx1, s);
        mx3 = fmaxf(mx3, s * s);                   // block-3 norm^2 = (|v|^2)^2
        float vv[3] = {v0, v1, v2};
        #pragma unroll
        for (int i = 0; i < 3; ++i)
            #pragma unroll
            for (int j = 0; j < 3; ++j)
                tp[(size_t)(m * 9 + i * 3 + j) * SP] = vv[i] * vv[j];
    }
    for (int m = 0; m < 4; ++m) {                  // block 2: (4,5)
        float s = 0.f;
        for (int k = 0; k < 5; ++k) {
            float v = x[(size_t)(64 + 5 * m + k) * SP];
            s += v * v;
        }
        mx2 = fmaxf(mx2, s);
    }
    atomicMax((int*)&factors[0], __float_as_int(mx0));
    atomicMax((int*)&factors[1], __float_as_int(mx1));
    atomicMax((int*)&factors[2], __float_as_int(mx2));
    atomicMax((int*)&factors[3], __float_as_int(mx3));
}

// ---------------------------------------------------------------------------
// Kernel 2: Wmat[96][228][392], packed tap tt = (kd*7+kh)*8 + kw
// rows 84..95 and kw==7 are zero; per-block norm factor folded into columns
// ---------------------------------------------------------------------------
__global__ void build_wmat(const float* __restrict__ basis,
                           const float* __restrict__ weights,
                           const unsigned* __restrict__ factors,
                           float* __restrict__ wmat) {
    long idx = (long)blockIdx.x * blockDim.x + threadIdx.x;
    const long total = (long)COUT_PAD * CIN * TAPS_PK;
    if (idx >= total) return;
    int tt  = (int)(idx % TAPS_PK);
    int col = (int)((idx / TAPS_PK) % CIN);
    int row = (int)(idx / ((long)TAPS_PK * CIN));
    int kw = tt & 7;
    int r  = tt >> 3;                 // kd*7 + kh
    if (row >= COUT || kw == 7) { wmat[idx] = 0.f; return; }
    int kd = r / 7, kh = r - kd * 7;
    int tap = kd * 49 + kh * 7 + kw;  // 0..342

    int o, u, a;
    if (row < 16)      { o = 0; u = row;          a = 0; }
    else if (row < 64) { o = 1; int q = row - 16; u = q / 3; a = q - 3 * u; }
    else               { o = 2; int q = row - 64; u = q / 5; a = q - 5 * u; }
    int i, v, bb;
    if (col < 16)      { i = 0; v = col;          bb = 0; }
    else if (col < 64) { i = 1; int q = col - 16; v = q / 3; bb = q - 3 * v; }
    else if (col < 84) { i = 2; int q = col - 64; v = q / 5; bb = q - 5 * v; }
    else               { i = 3; int q = col - 84; v = q / 9; bb = q - 9 * v; }

    const float* bas = basis   + ((((size_t)(o * 4 + i) * 9) * 5 + a) * 9 + bb) * TAPS + tap;
    const float* w   = weights + (((size_t)(o * 4 + i) * 9) * 16 + u) * 16 + v;
    float acc = 0.f;
    #pragma unroll
    for (int p = 0; p < 9; ++p)
        acc += w[(size_t)p * 256] * bas[(size_t)p * 5 * 9 * TAPS];

    float fac = sqrtf(__int_as_float((int)factors[i]) + 1e-12f);
    wmat[idx] = acc / (fac + 1e-5f);
}

// ---------------------------------------------------------------------------
// Kernel 3: implicit-GEMM conv via V_WMMA_F32_16X16X4_F32, double-buffered LDS
// block = 192 threads = 6 waves; block owns one (b,d,h) row of 32 voxels
// (two 16-wide N tiles); wave wv owns co-tile [wv*16, wv*16+16).
// Hot loop: 49 (kd,kh) rows x 2 K-slices, no divisions, no selects.
// One barrier per channel; next channel's loads overlap current WMMA work.
// ---------------------------------------------------------------------------
__device__ __forceinline__ const float* chan_src(const float* sv5, const float* tbuf,
                                                 int b, int c) {
    return (c < CIN_SV)
        ? sv5  + ((size_t)(b * CIN_SV + c) << 15)
        : tbuf + ((size_t)(b * CIN_T + (c - CIN_SV)) << 15);
}

__global__ __launch_bounds__(192)
void conv_wmma_kernel(const float* __restrict__ sv5,
                      const float* __restrict__ tbuf,
                      const float* __restrict__ wmat,
                      const float* __restrict__ bias,
                      float* __restrict__ out) {
    __shared__ float tile[2][TILE_N];

    int blk = blockIdx.x;            // NBATCH*32*32 = 2048
    int h = blk & 31;
    int d = (blk >> 5) & 31;
    int b = blk >> 10;
    int tid  = threadIdx.x;
    int lane = tid & 31;
    int wv   = tid >> 5;             // 0..5
    int co0  = wv << 4;
    int m16  = lane & 15;            // A row / B col
    int hi   = lane >> 4;            // K-half (A/B), M-half (C/D)

    // ---- precompute staging slots (invariant across channels) -------------
    int g_off[NSTAGE];
    #pragma unroll
    for (int q = 0; q < NSTAGE; ++q) {
        int s = tid + q * 192;
        int off = -1;
        if (s < TILE_N) {
            int dz  = s / (7 * TW);
            int rem = s - dz * (7 * TW);
            int dy  = rem / TW;
            int dx  = rem - dy * TW;
            int zz = d + dz - 3;
            int yy = h + dy - 3;
            int xx = dx - 3;
            if ((unsigned)zz < 32u && (unsigned)yy < 32u && (unsigned)xx < 32u)
                off = (zz << 10) + (yy << 5) + xx;
        }
        g_off[q] = off;
    }

    // ---- prologue: stage channel 0 into buffer 0 --------------------------
    {
        const float* src = chan_src(sv5, tbuf, b, 0);
        #pragma unroll
        for (int q = 0; q < NSTAGE; ++q) {
            int s = tid + q * 192;
            if (s < TILE_N)
                tile[0][s] = (g_off[q] >= 0) ? src[g_off[q]] : 0.f;
        }
    }
    __syncthreads();

    v8f acc0 = {};                   // n = 0..15
    v8f acc1 = {};                   // n = 16..31
    const int kA = hi << 1;          // lane's kw base within a K-slice

    for (int c = 0; c < CIN; ++c) {
        int p = c & 1;

        // ---- issue next channel's global loads early (hide under WMMA) ----
        float stg[NSTAGE];
        bool have_next = (c + 1 < CIN);
        if (have_next) {
            const float* srcn = chan_src(sv5, tbuf, b, c + 1);
            #pragma unroll
            for (int q = 0; q < NSTAGE; ++q) {
                int s = tid + q * 192;
                stg[q] = (s < TILE_N && g_off[q] >= 0) ? srcn[g_off[q]] : 0.f;
            }
        }

        // ---- compute current channel from buffer p ------------------------
        const float* wrow = wmat + ((size_t)(co0 + m16) * CIN + c) * TAPS_PK;
        __builtin_prefetch(wrow + 2 * TAPS_PK, 0, 1);   // weights two channels out
        const float* tbase = tile[p];

        #pragma unroll 7
        for (int r = 0; r < 49; ++r) {
            const float* tr = tbase + r * TW;
            const float* wr = wrow + (r << 3);
            // K-slice 0: kw 0..3
            v2f a0  = *(const v2f*)(wr + kA);
            v2f b00 = { tr[kA + m16],      tr[kA + 1 + m16] };
            v2f b01 = { tr[kA + 16 + m16], tr[kA + 17 + m16] };
            acc0 = __builtin_amdgcn_wmma_f32_16x16x4_f32(
                       false, a0, false, b00, (short)0, acc0, false, false);
            acc1 = __builtin_amdgcn_wmma_f32_16x16x4_f32(
                       false, a0, false, b01, (short)0, acc1, false, false);
            // K-slice 1: kw 4..7 (kw==7 weight is 0, B value irrelevant)
            v2f a1  = *(const v2f*)(wr + 4 + kA);
            v2f b10 = { tr[kA + 4 + m16],  tr[kA + 5 + m16] };
            v2f b11 = { tr[kA + 20 + m16], tr[kA + 21 + m16] };
            acc0 = __builtin_amdgcn_wmma_f32_16x16x4_f32(
                       false, a1, false, b10, (short)0, acc0, false, false);
            acc1 = __builtin_amdgcn_wmma_f32_16x16x4_f32(
                       false, a1, false, b11, (short)0, acc1, false, false);
        }

        // ---- commit staged registers into the other buffer ----------------
        if (have_next) {
            #pragma unroll
            for (int q = 0; q < NSTAGE; ++q) {
                int s = tid + q * 192;
                if (s < TILE_N)
                    tile[1 - p][s] = stg[q];
            }
        }
        __syncthreads();             // one barrier per channel
    }

    // C/D layout: VGPR j, lanes 0-15 -> M=j, lanes 16-31 -> M=j+8; N=lane&15
    int spbase = (d << 10) + (h << 5) + m16;
    #pragma unroll
    for (int j = 0; j < 8; ++j) {
        int co = co0 + j + (hi << 3);
        if (co < COUT) {
            float v0 = acc0[j];
            float v1 = acc1[j];
            if (co < 16) {
                float bb = bias[co];
                v0 = fmaxf(v0 + bb, 0.f);
                v1 = fmaxf(v1 + bb, 0.f);
            }
            size_t obase = ((size_t)(b * COUT + co) << 15);
            out[obase + spbase]      = v0;
            out[obase + spbase + 16] = v1;
        }
    }
}

// ---------------------------------------------------------------------------
extern "C" void kernel_launch(void* const* d_in, const int* in_sizes, int n_in,
                              void* d_out, int out_size, void* d_ws, size_t ws_size,
                              hipStream_t stream) {
    const float* sv5     = (const float*)d_in[0];
    const float* basis   = (const float*)d_in[1];
    const float* weights = (const float*)d_in[2];
    const float* bias    = (const float*)d_in[3];
    float* out = (float*)d_out;

    unsigned* factors = (unsigned*)d_ws;
    float* tbuf = (float*)((char*)d_ws + 256);
    size_t tbytes = (size_t)NBATCH * CIN_T * SP * sizeof(float);
    float* wmat = (float*)((char*)d_ws + 256 + tbytes);

    hipLaunchKernelGGL(init_factors, dim3(1), dim3(32), 0, stream, factors);
    hipLaunchKernelGGL(tensor_norm_kernel, dim3((NBATCH * SP) / 256), dim3(256),
                       0, stream, sv5, tbuf, factors);
    long total_w = (long)COUT_PAD * CIN * TAPS_PK;
    hipLaunchKernelGGL(build_wmat, dim3((unsigned)((total_w + 255) / 256)), dim3(256),
                       0, stream, basis, weights, factors, wmat);
    hipLaunchKernelGGL(conv_wmma_kernel, dim3(NBATCH * SDIM * SDIM), dim3(192),
                       0, stream, sv5, tbuf, wmat, bias, out);
}